// CalibrationCurve_51041391345836
// MI455X (gfx1250) — compile-verified
//
#include <hip/hip_runtime.h>

#define N_BINS    10
#define NROWS     30          // 3 stats x 10 bins; d_out layout: [prob(0-9), tp(10-19), count(20-29)]
#define NROWS_PAD 32          // pad to 2x16 for WMMA row groups
#define BLOCK     256
#define GRID      2048

typedef float v2f __attribute__((ext_vector_type(2)));
typedef float v4f __attribute__((ext_vector_type(4)));
typedef float v8f __attribute__((ext_vector_type(8)));

__global__ void calib_zero(float* __restrict__ out) {
    int t = threadIdx.x;
    if (t < NROWS) out[t] = 0.0f;
}

__global__ __launch_bounds__(BLOCK)
void calib_hist(const float* __restrict__ outputs,
                const float* __restrict__ labels,
                float* __restrict__ out, long long n) {
    __shared__ float hist[NROWS_PAD][BLOCK];
    const int t = threadIdx.x;

    // zero the per-column LDS histogram (incl. pad rows used by WMMA)
    #pragma unroll
    for (int r = 0; r < NROWS_PAD; ++r) hist[r][t] = 0.0f;
    __syncthreads();

    const float LOW0      = -1e-6f;
    const float STEP      = (1.0f - LOW0) / (float)N_BINS;
    const float INV_STEP  = 1.0f / STEP;
    const float HIGH_LAST = LOW0 + STEP * (float)N_BINS;   // ~1.0f

    const long long n4 = n >> 2;
    const v4f* __restrict__ o4 = (const v4f*)outputs;
    const v4f* __restrict__ l4 = (const v4f*)labels;
    const long long stride = (long long)gridDim.x * (long long)blockDim.x;

    for (long long i = (long long)blockIdx.x * blockDim.x + t; i < n4; i += stride) {
        v4f x = __builtin_nontemporal_load(&o4[i]);   // global_load_b128, TH=NT
        v4f y = __builtin_nontemporal_load(&l4[i]);
        #pragma unroll
        for (int e = 0; e < 4; ++e) {
            float xv = x[e], yv = y[e];
            bool valid = (xv > LOW0) & (xv <= HIGH_LAST);
            int b = (int)((xv - LOW0) * INV_STEP);
            b = min(max(b, 0), N_BINS - 1);
            if (valid) {
                float tp = (yv > 0.5f) ? 1.0f : 0.0f;
                // lane's LDS bank = tid mod 64 regardless of bin -> conflict-free ds_add_f32
                atomicAdd(&hist[b][t], xv);
                atomicAdd(&hist[N_BINS + b][t], tp);
                atomicAdd(&hist[2 * N_BINS + b][t], 1.0f);
            }
        }
    }

    // scalar tail (dead for n % 4 == 0, kept for generality)
    if (blockIdx.x == 0 && t == 0) {
        for (long long i = n4 << 2; i < n; ++i) {
            float xv = outputs[i], yv = labels[i];
            bool valid = (xv > LOW0) & (xv <= HIGH_LAST);
            int b = (int)((xv - LOW0) * INV_STEP);
            b = min(max(b, 0), N_BINS - 1);
            if (valid) {
                float tp = (yv > 0.5f) ? 1.0f : 0.0f;
                atomicAdd(&hist[b][0], xv);
                atomicAdd(&hist[N_BINS + b][0], tp);
                atomicAdd(&hist[2 * N_BINS + b][0], 1.0f);
            }
        }
    }
    __syncthreads();

    // WMMA epilogue: row-sum the 32x256 LDS histogram with D = A(16x4) * ones(4x16) + C.
    // Wave 0 -> rows 0..15, wave 1 -> rows 16..31. EXEC is all-ones inside each wave.
    const int wave = t >> 5;
    if (wave < 2) {
        const int lane  = t & 31;
        const int m     = lane & 15;          // A-matrix row for this lane
        const int khalf = (lane >> 4) << 1;   // lanes 0-15: K=0,1 ; lanes 16-31: K=2,3
        const int rbase = wave * 16;

        v8f acc = {};
        v2f bones; bones[0] = 1.0f; bones[1] = 1.0f;   // B = all-ones (layout-invariant)

        #pragma unroll 4
        for (int kc = 0; kc < BLOCK; kc += 4) {
            v2f a;
            a[0] = hist[rbase + m][kc + khalf + 0];
            a[1] = hist[rbase + m][kc + khalf + 1];
            acc = __builtin_amdgcn_wmma_f32_16x16x4_f32(
                /*neg_a=*/false, a, /*neg_b=*/false, bones,
                /*c_mod=*/(short)0, acc, /*reuse_a=*/false, /*reuse_b=*/false);
        }

        // D layout: VGPR r -> lanes 0-15: M=r (N=lane), lanes 16-31: M=8+r.
        // All N-columns of D are identical; lanes N==0 publish the row sums.
        if ((lane & 15) == 0) {
            const int mofs = (lane >> 4) * 8;
            #pragma unroll
            for (int r = 0; r < 8; ++r) {
                int row = rbase + mofs + r;
                if (row < NROWS) {
                    __hip_atomic_fetch_add(&out[row], acc[r], __ATOMIC_RELAXED,
                                           __HIP_MEMORY_SCOPE_AGENT);  // global_atomic_add_f32
                }
            }
        }
    }
}

extern "C" void kernel_launch(void* const* d_in, const int* in_sizes, int n_in,
                              void* d_out, int out_size, void* d_ws, size_t ws_size,
                              hipStream_t stream) {
    const float* outputs = (const float*)d_in[0];
    const float* labels  = (const float*)d_in[1];
    float* out = (float*)d_out;
    long long n = (long long)in_sizes[0];

    calib_zero<<<1, 32, 0, stream>>>(out);
    calib_hist<<<GRID, BLOCK, 0, stream>>>(outputs, labels, out, n);
}